// GraphSAGE_901943132494
// MI455X (gfx1250) — compile-verified
//
#include <hip/hip_runtime.h>
#include <hip/hip_bf16.h>

typedef __attribute__((ext_vector_type(2))) float v2f;
typedef __attribute__((ext_vector_type(8))) float v8f;

#define BN_EPS 1e-5f
#define NORM_EPS 1e-12f

// ---------------------------------------------------------------------------
// zero: grid-stride fill
// ---------------------------------------------------------------------------
__global__ __launch_bounds__(256) void zero_kernel(float* __restrict__ p, size_t n) {
    size_t i = (size_t)blockIdx.x * blockDim.x + threadIdx.x;
    size_t stride = (size_t)gridDim.x * blockDim.x;
    for (; i < n; i += stride) p[i] = 0.0f;
}

// ---------------------------------------------------------------------------
// scatter-mean accumulation: one wave (32 lanes) per edge, 4 floats/lane.
// Optionally applies BN(scale,shift)+ReLU to the gathered source feature.
// ---------------------------------------------------------------------------
template <bool BN>
__global__ __launch_bounds__(256) void scatter_kernel(
    const float* __restrict__ x, const int* __restrict__ src,
    const int* __restrict__ dst, float* __restrict__ agg,
    float* __restrict__ cnt, const float* __restrict__ st, int E) {
    int e = blockIdx.x * 8 + (threadIdx.x >> 5);
    if (e >= E) return;
    int lane = threadIdx.x & 31;
    int s = src[e];
    int d = dst[e];
    float4 v4 = ((const float4*)(x + (size_t)s * 128))[lane];
    float v[4] = {v4.x, v4.y, v4.z, v4.w};
    float* ad = agg + (size_t)d * 128 + lane * 4;
#pragma unroll
    for (int j = 0; j < 4; ++j) {
        float vv = v[j];
        if (BN) {
            int c = lane * 4 + j;
            vv = fmaxf(fmaf(vv, st[c], st[128 + c]), 0.0f);
        }
        atomicAdd(ad + j, vv);   // global_atomic_add_f32, no return
    }
    if (lane == 0) atomicAdd(&cnt[d], 1.0f);
}

// ---------------------------------------------------------------------------
// Fused SAGE GEMM: out = L2norm_rows( (agg/max(cnt,1)) @ Wl + bl + x @ Wr )
// plus per-column sum / sum-of-squares accumulation for the following BN.
// One block = 16 output rows; 8 waves each own a 16x16 output tile
// (wave w -> columns 16w..16w+15). Uses V_WMMA_F32_16X16X4_F32, K=128.
// Safe to call with out == x (blocks touch disjoint rows; reads precede
// writes within a block).
// ---------------------------------------------------------------------------
template <bool BN>
__global__ __launch_bounds__(256) void sage_gemm_kernel(
    const float* __restrict__ agg, const float* __restrict__ cnt,
    const float* __restrict__ x, const float* __restrict__ Wl,
    const float* __restrict__ bl, const float* __restrict__ Wr,
    float* __restrict__ out, float* __restrict__ colsum,
    float* __restrict__ colsumsq, const float* __restrict__ st, int N) {
    __shared__ float rowsq[16];

    const int tid  = threadIdx.x;
    const int wave = tid >> 5;       // 0..7
    const int lane = tid & 31;
    const int m    = lane & 15;      // A-fragment row
    const int hh   = lane >> 4;      // lane half (K split for A, M split for C)
    const int n    = wave * 16 + m;  // output column owned by this lane
    const int kh   = hh * 2;         // K offset within a 4-wide step
    const int row0 = blockIdx.x * 16;

    int rowm = row0 + m;
    if (rowm >= N) rowm = N - 1;     // clamp: keep EXEC all-1s for WMMA
    const float scl = 1.0f / fmaxf(cnt[rowm], 1.0f);
    const float* __restrict__ arow = agg + (size_t)rowm * 128;
    const float* __restrict__ xrow = x + (size_t)rowm * 128;

    v8f c = {};
#pragma unroll 4
    for (int k = 0; k < 128; k += 4) {
        const int ka = k + kh;
        v2f a, b;
        a.x = arow[ka] * scl;
        a.y = arow[ka + 1] * scl;
        b.x = Wl[ka * 128 + n];
        b.y = Wl[(ka + 1) * 128 + n];
        c = __builtin_amdgcn_wmma_f32_16x16x4_f32(false, a, false, b,
                                                  (short)0, c, false, false);
        float v0 = xrow[ka];
        float v1 = xrow[ka + 1];
        if (BN) {
            v0 = fmaxf(fmaf(v0, st[ka], st[128 + ka]), 0.0f);
            v1 = fmaxf(fmaf(v1, st[ka + 1], st[128 + ka + 1]), 0.0f);
        }
        v2f a2, b2;
        a2.x = v0;
        a2.y = v1;
        b2.x = Wr[ka * 128 + n];
        b2.y = Wr[(ka + 1) * 128 + n];
        c = __builtin_amdgcn_wmma_f32_16x16x4_f32(false, a2, false, b2,
                                                  (short)0, c, false, false);
    }

    // bias (lin_l bias only, per reference)
    const float bln = bl[n];
#pragma unroll
    for (int v = 0; v < 8; ++v) c[v] += bln;

    // Row L2 norm. C layout: lane holds column n, rows (8*hh + v), v=0..7.
    float sq[8];
#pragma unroll
    for (int v = 0; v < 8; ++v) sq[v] = c[v] * c[v];
#pragma unroll
    for (int off = 1; off < 16; off <<= 1) {
#pragma unroll
        for (int v = 0; v < 8; ++v) sq[v] += __shfl_xor(sq[v], off, 32);
    }
    // sq[v] now = wave-local (16 columns) sum of squares for row 8*hh+v
    if (tid < 16) rowsq[tid] = 0.0f;
    __syncthreads();
    if ((lane & 15) == 0) {
#pragma unroll
        for (int v = 0; v < 8; ++v) atomicAdd(&rowsq[hh * 8 + v], sq[v]);
    }
    __syncthreads();

    float psum = 0.0f, psumsq = 0.0f;
#pragma unroll
    for (int v = 0; v < 8; ++v) {
        const int r = row0 + hh * 8 + v;
        const float norm = sqrtf(rowsq[hh * 8 + v]);
        const float val = c[v] * (1.0f / fmaxf(norm, NORM_EPS));
        if (r < N) {
            out[(size_t)r * 128 + n] = val;
            psum += val;
            psumsq += val * val;
        }
    }
    // combine the two lane-halves (rows 0-7 with rows 8-15) per column
    psum += __shfl_xor(psum, 16, 32);
    psumsq += __shfl_xor(psumsq, 16, 32);
    if (hh == 0) {
        atomicAdd(&colsum[n], psum);
        atomicAdd(&colsumsq[n], psumsq);
    }
}

// ---------------------------------------------------------------------------
// BN stats -> per-column affine (scale, shift): y = max(0, x*scale + shift)
// ---------------------------------------------------------------------------
__global__ __launch_bounds__(128) void bn_stats_kernel(
    const float* __restrict__ colsum, const float* __restrict__ colsumsq,
    const float* __restrict__ g, const float* __restrict__ b, float invN,
    float* __restrict__ st) {
    int cidx = threadIdx.x;
    float mean = colsum[cidx] * invN;
    float var = colsumsq[cidx] * invN - mean * mean;
    float rs = rsqrtf(var + BN_EPS);
    float sc = g[cidx] * rs;
    st[cidx] = sc;
    st[128 + cidx] = b[cidx] - mean * sc;
}

// ---------------------------------------------------------------------------
// Predictor: one wave per label edge; BN2+ReLU applied on the fly.
// out[e] = dot(bn(x[s]), Wlin[0:128]) + dot(bn(x[d]), Wlin[128:256]) + blin
// ---------------------------------------------------------------------------
__global__ __launch_bounds__(256) void predict_kernel(
    const float* __restrict__ x, const float* __restrict__ st,
    const int* __restrict__ ls, const int* __restrict__ ld,
    const float* __restrict__ Wlin, const float* __restrict__ blin,
    float* __restrict__ out, int EL) {
    int e = blockIdx.x * 8 + (threadIdx.x >> 5);
    if (e >= EL) return;
    int lane = threadIdx.x & 31;
    int s = ls[e];
    int d = ld[e];
    float4 xs = ((const float4*)(x + (size_t)s * 128))[lane];
    float4 xd = ((const float4*)(x + (size_t)d * 128))[lane];
    float4 w0 = ((const float4*)Wlin)[lane];
    float4 w1 = ((const float4*)(Wlin + 128))[lane];
    float vs[4] = {xs.x, xs.y, xs.z, xs.w};
    float vd[4] = {xd.x, xd.y, xd.z, xd.w};
    float ws[4] = {w0.x, w0.y, w0.z, w0.w};
    float wd[4] = {w1.x, w1.y, w1.z, w1.w};
    float acc = 0.0f;
#pragma unroll
    for (int j = 0; j < 4; ++j) {
        int c = lane * 4 + j;
        float a = fmaxf(fmaf(vs[j], st[c], st[128 + c]), 0.0f);
        float b = fmaxf(fmaf(vd[j], st[c], st[128 + c]), 0.0f);
        acc = fmaf(a, ws[j], acc);
        acc = fmaf(b, wd[j], acc);
    }
#pragma unroll
    for (int off = 1; off < 32; off <<= 1) acc += __shfl_xor(acc, off, 32);
    if (lane == 0) out[e] = acc + blin[0];
}

// ---------------------------------------------------------------------------
// launch
// ---------------------------------------------------------------------------
extern "C" void kernel_launch(void* const* d_in, const int* in_sizes, int n_in,
                              void* d_out, int out_size, void* d_ws,
                              size_t ws_size, hipStream_t stream) {
    const int* ei   = (const int*)d_in[0];     // (2,E)
    const int* eli  = (const int*)d_in[1];     // (2,EL)
    const float* emb  = (const float*)d_in[2]; // (N,128)
    const float* Wl1  = (const float*)d_in[3];
    const float* bl1  = (const float*)d_in[4];
    const float* Wr1  = (const float*)d_in[5];
    const float* g1   = (const float*)d_in[6];
    const float* b1   = (const float*)d_in[7];
    const float* Wl2  = (const float*)d_in[8];
    const float* bl2  = (const float*)d_in[9];
    const float* Wr2  = (const float*)d_in[10];
    const float* g2   = (const float*)d_in[11];
    const float* b2   = (const float*)d_in[12];
    const float* Wlin = (const float*)d_in[13];
    const float* blin = (const float*)d_in[14];

    const int E  = in_sizes[0] / 2;
    const int EL = in_sizes[1] / 2;
    const int N  = in_sizes[2] / 128;

    const int* srcp = ei;
    const int* dstp = ei + E;
    const int* lsp  = eli;
    const int* ldp  = eli + EL;

    float* out = (float*)d_out;

    // workspace layout: A | B | cnt | colsum | colsumsq | stats1 | stats2
    char* w = (char*)d_ws;
    float* A = (float*)w;        w += (size_t)N * 128 * sizeof(float);
    float* B = (float*)w;        w += (size_t)N * 128 * sizeof(float);
    float* cnt = (float*)w;      w += (size_t)N * sizeof(float);
    float* colsum = (float*)w;   w += 128 * sizeof(float);
    float* colsumsq = (float*)w; w += 128 * sizeof(float);
    float* stats1 = (float*)w;   w += 256 * sizeof(float);
    float* stats2 = (float*)w;

    const size_t nF = (size_t)N * 128;
    const int zgridA = (int)((nF + 1023) / 1024);
    const int zgridC = (N + 256 + 255) / 256;  // cnt + colsum + colsumsq
    const int sgrid  = (E + 7) / 8;
    const int ggrid  = (N + 15) / 16;
    const int pgrid  = (EL + 7) / 8;
    const float invN = 1.0f / (float)N;

    // ---- layer 1 ----
    zero_kernel<<<zgridA, 256, 0, stream>>>(A, nF);
    zero_kernel<<<zgridC, 256, 0, stream>>>(cnt, (size_t)N + 256);
    scatter_kernel<false><<<sgrid, 256, 0, stream>>>(emb, srcp, dstp, A, cnt,
                                                     nullptr, E);
    sage_gemm_kernel<false><<<ggrid, 256, 0, stream>>>(
        A, cnt, emb, Wl1, bl1, Wr1, B, colsum, colsumsq, nullptr, N);
    bn_stats_kernel<<<1, 128, 0, stream>>>(colsum, colsumsq, g1, b1, invN,
                                           stats1);

    // ---- layer 2 (BN1+ReLU applied on the fly; GEMM writes B in place) ----
    zero_kernel<<<zgridA, 256, 0, stream>>>(A, nF);
    zero_kernel<<<zgridC, 256, 0, stream>>>(cnt, (size_t)N + 256);
    scatter_kernel<true><<<sgrid, 256, 0, stream>>>(B, srcp, dstp, A, cnt,
                                                    stats1, E);
    sage_gemm_kernel<true><<<ggrid, 256, 0, stream>>>(
        A, cnt, B, Wl2, bl2, Wr2, B, colsum, colsumsq, stats1, N);
    bn_stats_kernel<<<1, 128, 0, stream>>>(colsum, colsumsq, g2, b2, invN,
                                           stats2);

    // ---- predictor (BN2+ReLU on the fly) ----
    predict_kernel<<<pgrid, 256, 0, stream>>>(B, stats2, lsp, ldp, Wlin, blin,
                                              out, EL);
}